// FieldAwareFactorizationMachine_28991029248073
// MI455X (gfx1250) — compile-verified
//
#include <hip/hip_runtime.h>
#include <hip/hip_bf16.h>
#include <stdint.h>

// FFM pairwise interactions, MI455X (gfx1250).
// out[b,p,d] = W[j, x[b,i]+i*FDIM, d] * W[i, x[b,j]+j*FDIM, d]  for pair p=(i<j)
//
// Memory-bound gather op (~755 MB traffic, 1 FLOP/element => ~32us floor @23.3TB/s).
// CDNA5 async data path: global_load_async_to_lds_b128 stages the 240 gathered
// embedding vectors of a sample into LDS (tracked by ASYNCcnt, drained with
// s_wait_asynccnt), then a vectorized float4 multiply/store phase streams
// coalesced b128 stores out of LDS. No matmul structure exists -> no WMMA.

#define NF    16
#define FDIM  31250
#define VOCAB 500000      // 16 * 31250
#define ED    32
#define NPAIR 120

// nibble-packed (i<<4)|j for the 120 upper-triangular pairs, row-major (k=1)
__constant__ unsigned char cPair[NPAIR] = {
  // i=0, j=1..15
    1,  2,  3,  4,  5,  6,  7,  8,  9, 10, 11, 12, 13, 14, 15,
  // i=1, j=2..15
   18, 19, 20, 21, 22, 23, 24, 25, 26, 27, 28, 29, 30, 31,
  // i=2
   35, 36, 37, 38, 39, 40, 41, 42, 43, 44, 45, 46, 47,
  // i=3
   52, 53, 54, 55, 56, 57, 58, 59, 60, 61, 62, 63,
  // i=4
   69, 70, 71, 72, 73, 74, 75, 76, 77, 78, 79,
  // i=5
   86, 87, 88, 89, 90, 91, 92, 93, 94, 95,
  // i=6
  103,104,105,106,107,108,109,110,111,
  // i=7
  120,121,122,123,124,125,126,127,
  // i=8
  137,138,139,140,141,142,143,
  // i=9
  154,155,156,157,158,159,
  // i=10
  171,172,173,174,175,
  // i=11
  188,189,190,191,
  // i=12
  205,206,207,
  // i=13
  222,223,
  // i=14
  239
};

__global__ __launch_bounds__(256) void
FieldAwareFactorizationMachine_28991029248073_kernel(const int* __restrict__ x,
                                                     const float* __restrict__ W,
                                                     float* __restrict__ out)
{
    // slot (src*16+dst) holds W[dst, gidx(src), 0:32]  (diagonal slots unused)
    __shared__ __align__(16) float sVec[NF * NF * ED];   // 32 KB
    __shared__ int sIdx[NF];

    const int b = blockIdx.x;
    const int t = threadIdx.x;

    if (t < NF) {
        sIdx[t] = x[b * NF + t] + t * FDIM;              // global vocab index per field
    }
    __syncthreads();

    // ---- async gather phase: 240 vectors * 128B, as 16B per lane-op (1920 ops) ----
    const unsigned           ldsBase = (unsigned)(uintptr_t)(&sVec[0]);
    const unsigned long long wBase   = (unsigned long long)(uintptr_t)W;

    for (int l = t; l < 240 * 8; l += 256) {
        const int o   = l >> 3;                          // ordered pair 0..239
        const int q   = l & 7;                           // float4 chunk 0..7
        const int src = o / 15;
        const int k   = o - src * 15;
        const int dst = k + (k >= src);

        const unsigned long long v = (unsigned long long)sIdx[src];   // 0..499999
        const unsigned long long gaddr =
            wBase + (((unsigned long long)dst * VOCAB + v) * ED + (unsigned)(q * 4)) * 4ull;
        const unsigned ldsAddr =
            ldsBase + (unsigned)(((src * NF + dst) * ED + q * 4) * 4);

        // CDNA5 async copy: copies 16 bytes from global address gaddr to LDS address ldsAddr, tracked by ASYNCcnt
        asm volatile("global_load_async_to_lds_b128 %0, %1, off"
                     :: "v"(ldsAddr), "v"(gaddr)
                     : "memory");
    }

    asm volatile("s_wait_asynccnt 0" ::: "memory");      // my wave's async ops done
    __syncthreads();                                     // everyone's LDS data visible

    // ---- compute phase: 120 pairs * 32 dims = 960 float4 groups, coalesced b128 ----
    float4* __restrict__ outB4 = (float4*)(out + (size_t)b * (NPAIR * ED));

    #pragma unroll 4
    for (int g = t; g < NPAIR * (ED / 4); g += 256) {
        const int p  = g >> 3;                           // pair index
        const int q  = g & 7;                            // float4 chunk within pair
        const int s  = cPair[p];                         // (i<<4)|j
        const int sw = ((s & 15) << 4) | (s >> 4);       // (j<<4)|i

        const float4 a = *(const float4*)&sVec[s  * ED + q * 4];  // W[j, idx_i, 4q..]
        const float4 c = *(const float4*)&sVec[sw * ED + q * 4];  // W[i, idx_j, 4q..]

        float4 r;
        r.x = a.x * c.x;
        r.y = a.y * c.y;
        r.z = a.z * c.z;
        r.w = a.w * c.w;
        outB4[g] = r;                                    // global_store_b128, unit-stride
    }
}

extern "C" void kernel_launch(void* const* d_in, const int* in_sizes, int n_in,
                              void* d_out, int out_size, void* d_ws, size_t ws_size,
                              hipStream_t stream) {
    const int*   x = (const int*)d_in[0];    // (BATCH, 16) int32
    const float* W = (const float*)d_in[1];  // (16, 500000, 32) float32
    float*     out = (float*)d_out;          // (BATCH, 120, 32) float32

    const int batch = in_sizes[0] / NF;      // 16384
    FieldAwareFactorizationMachine_28991029248073_kernel<<<batch, 256, 0, stream>>>(x, W, out);
}